// RMILoss_8735963480104
// MI455X (gfx1250) — compile-verified
//
#include <hip/hip_runtime.h>
#include <hip/hip_bf16.h>
#include <math.h>

// ---------------------------------------------------------------------------
// RMI loss for MI455X (gfx1250, wave32).
//  pred/target: (8, 8, 512, 512) f32.  Output: scalar f32.
//  Pass 1: fused sigmoid + BCE + 3x3/s3 avg-pool  (HBM-bound, single pass,
//          1 exp + 1 log + 1 rcp per element)
//  Pass 2: 9x9 Gram matrices via V_WMMA_F32_16X16X4_F32, branch-free loop
//          (pad lanes load clamped data that only lands in unread rows/cols)
//  Pass 3: per-(b,c) 9x9 solve + logdet
//  Pass 4: finalize scalar
// ---------------------------------------------------------------------------

typedef float v2f __attribute__((ext_vector_type(2)));
typedef float v8f __attribute__((ext_vector_type(8)));

#define BC      64          // 8 batches * 8 channels
#define IMG     512
#define PW      170         // pooled width  (512/3)
#define NW      168         // region grid   (170-2)
#define NS      (NW*NW)     // samples per (b,c) = 28224
#define ROWS_PER_WAVE 21    // 168 rows / 8 waves
#define EPSI    0.0005f

// ---------------------------------------------------------------- utilities
// p = sigmoid(x);  bce term = t*sp(-x) + (1-t)*sp(x) = softplus(x) - t*x
__device__ __forceinline__ void sig_bce(float x, float t, float& p, float& bce) {
    const float ax = fabsf(x);
    const float u  = __expf(-ax);           // in (0, 1]
    const float l  = __logf(1.f + u);       // == log1p(u), arg in (1,2]
    bce += fmaxf(x, 0.f) + l - t * x;       // softplus(x) - t*x
    const float s = __frcp_rn(1.f + u);     // sigmoid(|x|)
    p = (x >= 0.f) ? s : 1.f - s;
}

// ------------------------------------------------------------------ kernel 0
__global__ void rmi_init_kernel(float* accs) {
    if (threadIdx.x < 2) accs[threadIdx.x] = 0.f;   // [0]=bce, [1]=rmi
}

// ------------------------------------------------------------------ kernel 1
// One block per (b,c) image. Fused: sigmoid, BCE partial sum, 3x3 avg-pool.
__global__ void pool_bce_kernel(const float* __restrict__ pred,
                                const float* __restrict__ target,
                                float* __restrict__ pd,
                                float* __restrict__ td,
                                float* __restrict__ accs) {
    const int bc = blockIdx.x;
    const float* X = pred   + (size_t)bc * IMG * IMG;
    const float* T = target + (size_t)bc * IMG * IMG;
    float* PD = pd + (size_t)bc * PW * PW;
    float* TD = td + (size_t)bc * PW * PW;

    float bce = 0.f;

    // pooled windows cover rows/cols [0, 510)
    for (int o = threadIdx.x; o < PW * PW; o += blockDim.x) {
        const int oi = o / PW;
        const int oj = o - oi * PW;
        const int base = (3 * oi) * IMG + 3 * oj;
        float sp = 0.f, st = 0.f;
        #pragma unroll
        for (int dy = 0; dy < 3; ++dy) {
            #pragma unroll
            for (int dx = 0; dx < 3; ++dx) {
                const int off = base + dy * IMG + dx;
                const float x = X[off];
                const float t = T[off];
                float p;
                sig_bce(x, t, p, bce);
                sp += p;
                st += t;
            }
        }
        PD[o] = sp * (1.f / 9.f);
        TD[o] = st * (1.f / 9.f);
    }

    // border elements not covered by pooling: rows 510..511 (full) plus
    // cols 510..511 for rows 0..509  -> 2*512 + 2*510 = 2044 elements
    for (int e = threadIdx.x; e < 2044; e += blockDim.x) {
        int row, col;
        if (e < 1024) { row = 510 + (e >> 9); col = e & 511; }
        else          { int e2 = e - 1024; row = e2 >> 1; col = 510 + (e2 & 1); }
        const int off = row * IMG + col;
        float p;
        sig_bce(X[off], T[off], p, bce);
    }

    // block reduction
    __shared__ float sb[256];
    sb[threadIdx.x] = bce;
    __syncthreads();
    for (int s = 128; s > 0; s >>= 1) {
        if (threadIdx.x < s) sb[threadIdx.x] += sb[threadIdx.x + s];
        __syncthreads();
    }
    if (threadIdx.x == 0) atomicAdd(&accs[0], sb[0]);
}

// ------------------------------------------------------------------ kernel 2
// One block (8 waves) per (b,c).  Each wave accumulates partial 16x16 f32
// Grams with V_WMMA_F32_16X16X4_F32.  For X*X^T the A-layout and B-layout of
// 16x16x4 f32 WMMA are identical, so the same VGPR pair feeds A and B.
// Branch-free: lanes with comp>=9 load clamped (comp=8) data; their values
// land only in D rows M>=9 / cols N>=9, which are never read.  Each wave
// owns exactly 21 sample rows (3528 samples), 42 K-chunks of 4 per row.
__global__ void gram_kernel(const float* __restrict__ pd,
                            const float* __restrict__ td,
                            float* __restrict__ grams,   // [BC][3][81]
                            float* __restrict__ means) { // [BC][18]
    const int bc = blockIdx.x;
    const float* Y = td + (size_t)bc * PW * PW;   // y = pooled target
    const float* P = pd + (size_t)bc * PW * PW;   // p = pooled sigmoid(pred)

    const int tid  = threadIdx.x;      // 0..255
    const int wave = tid >> 5;         // 0..7
    const int lane = tid & 31;
    const int comp = lane & 15;        // region component (valid < 9)
    const int lh   = lane >> 4;        // lane half: samples +0/+1 vs +2/+3
    const int cc   = (comp < 9) ? comp : 8;   // clamp pad lanes to real data
    const int cy   = cc / 3;
    const int cx   = cc - 3 * cy;

    // per-lane base: sample row i0 = wave*21, component offset, K-half offset
    const int i0   = wave * ROWS_PER_WAVE;
    const int lofs = (i0 + cy) * PW + cx + 2 * lh;
    const float* Yb = Y + lofs;
    const float* Pb = P + lofs;

    v8f accY = {}, accP = {}, accC = {};
    float sumY = 0.f, sumP = 0.f;

    for (int ii = 0; ii < ROWS_PER_WAVE; ++ii) {
        const int rbase = ii * PW;
        #pragma unroll 2
        for (int jc = 0; jc < NW / 4; ++jc) {       // 42 K-chunks per row
            const int off = rbase + jc * 4;
            const float y0 = Yb[off];
            const float y1 = Yb[off + 1];
            const float p0 = Pb[off];
            const float p1 = Pb[off + 1];
            const v2f av = { y0, y1 };   // A/B tile of y
            const v2f bv = { p0, p1 };   // A/B tile of p
            accY = __builtin_amdgcn_wmma_f32_16x16x4_f32(false, av, false, av,
                                                         (short)0, accY, false, false);
            accP = __builtin_amdgcn_wmma_f32_16x16x4_f32(false, bv, false, bv,
                                                         (short)0, accP, false, false);
            accC = __builtin_amdgcn_wmma_f32_16x16x4_f32(false, av, false, bv,
                                                         (short)0, accC, false, false);
            sumY += y0 + y1;
            sumP += p0 + p1;
        }
    }

    // -------- cross-wave reduction through LDS, one matrix at a time -------
    __shared__ float red[8 * 32 * 8];     // 8 KB
    __shared__ float msum[256 * 2];

    v8f accs3[3] = { accY, accP, accC };
    #pragma unroll
    for (int mat = 0; mat < 3; ++mat) {
        #pragma unroll
        for (int v = 0; v < 8; ++v) red[wave * 256 + lane * 8 + v] = accs3[mat][v];
        __syncthreads();
        if (wave == 0) {
            v8f f = {};
            for (int w = 0; w < 8; ++w) {
                #pragma unroll
                for (int v = 0; v < 8; ++v) f[v] += red[w * 256 + lane * 8 + v];
            }
            // C/D layout: element (M,N): N = lane&15, M = v + 8*(lane>>4)
            #pragma unroll
            for (int v = 0; v < 8; ++v) {
                const int M = v + 8 * lh;
                const int N = comp;
                if (M < 9 && N < 9)
                    grams[(size_t)bc * 243 + mat * 81 + M * 9 + N] = f[v];
            }
        }
        __syncthreads();
    }

    // component sums -> means (pad-lane slots t&15 >= 9 are never read)
    msum[tid * 2 + 0] = sumY;
    msum[tid * 2 + 1] = sumP;
    __syncthreads();
    if (tid < 9) {
        float a = 0.f, b = 0.f;
        for (int g = 0; g < 16; ++g) {          // threads with (t&15)==tid
            const int t = tid + 16 * g;
            a += msum[t * 2 + 0];
            b += msum[t * 2 + 1];
        }
        means[bc * 18 + tid]     = a / (float)NS;   // my
        means[bc * 18 + 9 + tid] = b / (float)NS;   // mp
    }
}

// ------------------------------------------------------------------ kernel 3
// 64 threads: per-(b,c) 9x9 covariance solve + logdet.
__global__ void solve_kernel(const float* __restrict__ grams,
                             const float* __restrict__ means,
                             float* __restrict__ accs) {
    const int bc = threadIdx.x;
    if (bc >= BC) return;
    const float* G  = grams + (size_t)bc * 243;
    const float* my = means + bc * 18;
    const float* mp = my + 9;
    const float n = (float)NS;

    float Yc[9][9], Cc[9][9];
    float A[9][18];   // [Pe | C^T] augmented for Gauss-Jordan
    for (int r = 0; r < 9; ++r)
        for (int s = 0; s < 9; ++s) {
            Yc[r][s] = G[0 * 81 + r * 9 + s] - n * my[r] * my[s];
            A[r][s]  = G[1 * 81 + r * 9 + s] - n * mp[r] * mp[s]
                       + (r == s ? EPSI : 0.f);
            Cc[r][s] = G[2 * 81 + r * 9 + s] - n * my[r] * mp[s];
        }
    for (int r = 0; r < 9; ++r)
        for (int s = 0; s < 9; ++s)
            A[r][9 + s] = Cc[s][r];           // RHS = C^T

    // Gauss-Jordan with partial pivoting: A[:,9:] becomes inv(Pe)*C^T
    for (int c = 0; c < 9; ++c) {
        int piv = c; float mx = fabsf(A[c][c]);
        for (int r = c + 1; r < 9; ++r)
            if (fabsf(A[r][c]) > mx) { mx = fabsf(A[r][c]); piv = r; }
        if (piv != c)
            for (int j = 0; j < 18; ++j) { float t = A[c][j]; A[c][j] = A[piv][j]; A[piv][j] = t; }
        const float inv = 1.f / A[c][c];
        for (int j = 0; j < 18; ++j) A[c][j] *= inv;
        for (int r = 0; r < 9; ++r) {
            if (r == c) continue;
            const float f = A[r][c];
            if (f != 0.f)
                for (int j = 0; j < 18; ++j) A[r][j] -= f * A[c][j];
        }
    }

    // m = Yc - C * X  (+ eps*I), then log|det| via LU
    float M[9][9];
    for (int r = 0; r < 9; ++r)
        for (int s = 0; s < 9; ++s) {
            float acc = Yc[r][s];
            for (int k = 0; k < 9; ++k) acc -= Cc[r][k] * A[k][9 + s];
            M[r][s] = acc + (r == s ? EPSI : 0.f);
        }

    float ld = 0.f;
    for (int c = 0; c < 9; ++c) {
        int piv = c; float mx = fabsf(M[c][c]);
        for (int r = c + 1; r < 9; ++r)
            if (fabsf(M[r][c]) > mx) { mx = fabsf(M[r][c]); piv = r; }
        if (piv != c)
            for (int j = c; j < 9; ++j) { float t = M[c][j]; M[c][j] = M[piv][j]; M[piv][j] = t; }
        const float d = M[c][c];
        ld += __logf(fabsf(d));
        const float inv = 1.f / d;
        for (int r = c + 1; r < 9; ++r) {
            const float f = M[r][c] * inv;
            for (int j = c; j < 9; ++j) M[r][j] -= f * M[c][j];
        }
    }

    // rmi_bc = 0.5 * (2*logdet) / 9
    atomicAdd(&accs[1], ld * (1.f / 9.f));
}

// ------------------------------------------------------------------ kernel 4
__global__ void finalize_kernel(const float* __restrict__ accs,
                                float* __restrict__ out) {
    if (threadIdx.x == 0) {
        const float bce = accs[0] / (float)(BC * (size_t)IMG * IMG);
        const float rmi = accs[1] / 8.f;         // sum over c, mean over b
        out[0] = 0.5f * rmi + 0.5f * bce;
    }
}

// ---------------------------------------------------------------------------
extern "C" void kernel_launch(void* const* d_in, const int* in_sizes, int n_in,
                              void* d_out, int out_size, void* d_ws, size_t ws_size,
                              hipStream_t stream) {
    const float* pred   = (const float*)d_in[0];
    const float* target = (const float*)d_in[1];
    float* out = (float*)d_out;

    // workspace layout (floats)
    float* ws    = (float*)d_ws;
    float* pd    = ws;                               // 64 * 170*170
    float* td    = pd    + (size_t)BC * PW * PW;     // 64 * 170*170
    float* grams = td    + (size_t)BC * PW * PW;     // 64 * 243
    float* means = grams + (size_t)BC * 243;         // 64 * 18
    float* accs  = means + (size_t)BC * 18;          // [bce, rmi]

    rmi_init_kernel<<<1, 32, 0, stream>>>(accs);
    pool_bce_kernel<<<BC, 256, 0, stream>>>(pred, target, pd, td, accs);
    gram_kernel<<<BC, 256, 0, stream>>>(pd, td, grams, means);
    solve_kernel<<<1, 64, 0, stream>>>(grams, means, accs);
    finalize_kernel<<<1, 32, 0, stream>>>(accs, out);
}